// ForwardKinematics_893353198094
// MI455X (gfx1250) — compile-verified
//
#include <hip/hip_runtime.h>

// Forward kinematics (hand model) — memory-bound streaming kernel.
// d_out layout: abs_xyz (B,21,3) flat, then uv (B,21,2) flat.
// Roofline: 656 B/element * 262144 = 172 MB  =>  ~7.4 us floor @ 23.3 TB/s.
// Specialized math (26 sincos pairs + ~1.3 kFLOP/element) keeps VALU well
// below that floor. WMMA intentionally unused: all 3x3 operands are
// per-batch-element, so no shared operand exists for a 16x16 tile.

__device__ __forceinline__ void rot_xyz_full(float ax, float ay, float az, float R[3][3]) {
    float sx, cx, sy, cy, sz, cz;
    __sincosf(ax, &sx, &cx);
    __sincosf(ay, &sy, &cy);
    __sincosf(az, &sz, &cz);
    // Rx*Ry*Rz closed form
    R[0][0] = cy * cz;
    R[0][1] = -cy * sz;
    R[0][2] = sy;
    R[1][0] = fmaf(sx * sy, cz, cx * sz);
    R[1][1] = fmaf(-(sx * sy), sz, cx * cz);
    R[1][2] = -sx * cy;
    R[2][0] = fmaf(-(cx * sy), cz, sx * sz);
    R[2][1] = fmaf(cx * sy, sz, sx * cz);
    R[2][2] = cx * cy;
}

__device__ __forceinline__ void matmul3(const float A[3][3], const float Bm[3][3], float C[3][3]) {
#pragma unroll
    for (int r = 0; r < 3; ++r) {
#pragma unroll
        for (int c = 0; c < 3; ++c) {
            C[r][c] = fmaf(A[r][0], Bm[0][c], fmaf(A[r][1], Bm[1][c], A[r][2] * Bm[2][c]));
        }
    }
}

// R = R @ Rx(th): col1' = c*col1 + s*col2 ; col2' = -s*col1 + c*col2
__device__ __forceinline__ void rmul_Rx(float R[3][3], float th) {
    float s, c;
    __sincosf(th, &s, &c);
#pragma unroll
    for (int r = 0; r < 3; ++r) {
        float a = R[r][1], b = R[r][2];
        R[r][1] = fmaf(a, c, b * s);
        R[r][2] = fmaf(b, c, -(a * s));
    }
}

// R = R @ Ry(th): col0' = c*col0 - s*col2 ; col2' = s*col0 + c*col2
__device__ __forceinline__ void rmul_Ry(float R[3][3], float th) {
    float s, c;
    __sincosf(th, &s, &c);
#pragma unroll
    for (int r = 0; r < 3; ++r) {
        float a = R[r][0], b = R[r][2];
        R[r][0] = fmaf(a, c, -(b * s));
        R[r][2] = fmaf(b, c, a * s);
    }
}

// Scale+translate point, store xyz (merged b96), project through K, store uv (merged b64).
__device__ __forceinline__ void emit_point(float* __restrict__ oxyz, float* __restrict__ ouv,
                                           int n, float px, float py, float pz,
                                           float irbl, float rx, float ry, float rz,
                                           const float K[3][3]) {
    float X = fmaf(px, irbl, rx);
    float Y = fmaf(py, irbl, ry);
    float Z = fmaf(pz, irbl, rz);
    oxyz[n * 3 + 0] = X;
    oxyz[n * 3 + 1] = Y;
    oxyz[n * 3 + 2] = Z;
    float q0 = fmaf(K[0][0], X, fmaf(K[0][1], Y, K[0][2] * Z));
    float q1 = fmaf(K[1][0], X, fmaf(K[1][1], Y, K[1][2] * Z));
    float q2 = fmaf(K[2][0], X, fmaf(K[2][1], Y, K[2][2] * Z));
    float z  = (q2 == 0.0f) ? 1e-10f : q2;
    float inv = __builtin_amdgcn_rcpf(z);   // v_rcp_f32 (TRANS, co-executes with VALU)
    ouv[n * 2 + 0] = q0 * inv;
    ouv[n * 2 + 1] = q1 * inv;
}

__global__ __launch_bounds__(256) void ForwardKinematics_893353198094_kernel(
    const float* __restrict__ root_angles,   // (B,3)
    const float* __restrict__ other_angles,  // (B,23)
    const float* __restrict__ bone_lengths,  // (B,20) rows 16B-aligned (stride 80B)
    const float* __restrict__ cam,           // (B,3,3)
    const float* __restrict__ irbl_in,       // (B,1)
    const float* __restrict__ kp_root,       // (B,3)
    float* __restrict__ out,
    unsigned nB)
{
    unsigned i = blockIdx.x * blockDim.x + threadIdx.x;
    if (i >= nB) return;

    // Unsigned flat index -> zero-extended 64-bit address math (cheaper than sext).
    const float* oa = other_angles + (size_t)i * 23u;
    const float* cm = cam + (size_t)i * 9u;
    const float* bl = bone_lengths + (size_t)i * 20u;

    // Pull this thread's scattered input rows toward the WGP before the
    // sincos dependency chains begin (global_prefetch_b8, speculative).
    __builtin_prefetch(oa, 0, 1);
    __builtin_prefetch(oa + 22, 0, 1);
    __builtin_prefetch(cm, 0, 1);
    __builtin_prefetch(bl, 0, 1);

    float* oxyz = out + (size_t)i * 63u;
    float* ouv  = out + (size_t)nB * 63u + (size_t)i * 42u;

    float R_root[3][3];
    rot_xyz_full(root_angles[(size_t)i * 3u + 0u],
                 root_angles[(size_t)i * 3u + 1u],
                 root_angles[(size_t)i * 3u + 2u], R_root);

    float K[3][3];
#pragma unroll
    for (int r = 0; r < 3; ++r)
#pragma unroll
        for (int c = 0; c < 3; ++c)
            K[r][c] = cm[r * 3 + c];

    float irbl = irbl_in[i];
    float rx = kp_root[(size_t)i * 3u + 0u];
    float ry = kp_root[(size_t)i * 3u + 1u];
    float rz = kp_root[(size_t)i * 3u + 2u];

    // Point 0 (wrist): local pos = 0 -> abs = kp_root
    emit_point(oxyz, ouv, 0, 0.f, 0.f, 0.f, irbl, rx, ry, rz, K);

    const float4* bl4 = reinterpret_cast<const float4*>(bl);  // 16B-aligned rows

    // ---- Finger 0: thumb (full rot, full rot, Ry, identity) ----
    {
        float4 bone = bl4[0];
        float L[3][3], Ra[3][3], Rb[3][3];

        rot_xyz_full(oa[0], oa[1], oa[2], L);
        matmul3(R_root, L, Ra);
        float px = bone.x * Ra[0][2], py = bone.x * Ra[1][2], pz = bone.x * Ra[2][2];
        emit_point(oxyz, ouv, 4, px, py, pz, irbl, rx, ry, rz, K);   // level 0 -> n=4

        rot_xyz_full(oa[3], oa[4], oa[5], L);
        matmul3(Ra, L, Rb);
        px = fmaf(bone.y, Rb[0][2], px); py = fmaf(bone.y, Rb[1][2], py); pz = fmaf(bone.y, Rb[2][2], pz);
        emit_point(oxyz, ouv, 3, px, py, pz, irbl, rx, ry, rz, K);   // level 1 -> n=3

        rmul_Ry(Rb, oa[6]);
        px = fmaf(bone.z, Rb[0][2], px); py = fmaf(bone.z, Rb[1][2], py); pz = fmaf(bone.z, Rb[2][2], pz);
        emit_point(oxyz, ouv, 2, px, py, pz, irbl, rx, ry, rz, K);   // level 2 -> n=2

        px = fmaf(bone.w, Rb[0][2], px); py = fmaf(bone.w, Rb[1][2], py); pz = fmaf(bone.w, Rb[2][2], pz);
        emit_point(oxyz, ouv, 1, px, py, pz, irbl, rx, ry, rz, K);   // level 3 -> n=1
    }

    // ---- Fingers 1..4: Rx*Ry, Rx, Rx, identity ----
#pragma unroll
    for (int f = 1; f <= 4; ++f) {
        float4 bone = bl4[f];
        const float* w = oa + 7 + (f - 1) * 4;

        float sx, cx, sy, cy;
        __sincosf(w[0], &sx, &cx);
        __sincosf(w[1], &sy, &cy);
        float L[3][3];
        L[0][0] = cy;       L[0][1] = 0.0f; L[0][2] = sy;
        L[1][0] = sx * sy;  L[1][1] = cx;   L[1][2] = -sx * cy;
        L[2][0] = -cx * sy; L[2][1] = sx;   L[2][2] = cx * cy;

        float R[3][3];
        matmul3(R_root, L, R);

        int base = 1 + f * 4;
        float px = bone.x * R[0][2], py = bone.x * R[1][2], pz = bone.x * R[2][2];
        emit_point(oxyz, ouv, base + 3, px, py, pz, irbl, rx, ry, rz, K);  // level 0

        rmul_Rx(R, w[2]);
        px = fmaf(bone.y, R[0][2], px); py = fmaf(bone.y, R[1][2], py); pz = fmaf(bone.y, R[2][2], pz);
        emit_point(oxyz, ouv, base + 2, px, py, pz, irbl, rx, ry, rz, K);  // level 1

        rmul_Rx(R, w[3]);
        px = fmaf(bone.z, R[0][2], px); py = fmaf(bone.z, R[1][2], py); pz = fmaf(bone.z, R[2][2], pz);
        emit_point(oxyz, ouv, base + 1, px, py, pz, irbl, rx, ry, rz, K);  // level 2

        px = fmaf(bone.w, R[0][2], px); py = fmaf(bone.w, R[1][2], py); pz = fmaf(bone.w, R[2][2], pz);
        emit_point(oxyz, ouv, base + 0, px, py, pz, irbl, rx, ry, rz, K);  // level 3
    }
}

extern "C" void kernel_launch(void* const* d_in, const int* in_sizes, int n_in,
                              void* d_out, int out_size, void* d_ws, size_t ws_size,
                              hipStream_t stream) {
    (void)n_in; (void)out_size; (void)d_ws; (void)ws_size;
    const float* root_angles  = (const float*)d_in[0];
    const float* other_angles = (const float*)d_in[1];
    const float* bone_lengths = (const float*)d_in[2];
    const float* cam          = (const float*)d_in[3];
    const float* irbl         = (const float*)d_in[4];
    const float* kp_root      = (const float*)d_in[5];
    float* out = (float*)d_out;

    unsigned nB = (unsigned)(in_sizes[0] / 3);   // root_angles is (B,3)
    unsigned block = 256;                        // 8 wave32 per block
    unsigned grid = (nB + block - 1) / block;

    hipLaunchKernelGGL(ForwardKinematics_893353198094_kernel,
                       dim3(grid), dim3(block), 0, stream,
                       root_angles, other_angles, bone_lengths, cam, irbl, kp_root,
                       out, nB);
}